// MyModel_55448027791612
// MI455X (gfx1250) — compile-verified
//
#include <hip/hip_runtime.h>
#include <hip/hip_bf16.h>
#include <math.h>

typedef __attribute__((ext_vector_type(2))) float v2f;
typedef __attribute__((ext_vector_type(8))) float v8f;

#define C_     144
#define W_     9225
#define KW_    10
#define F_     64
#define H1_    128
#define CAT_   18432
#define TCONV_ 9216      // W - KW + 1
#define NCHUNK 2306      // full K=4 chunks in W (4*2306 = 9224, leftover w=9224)
#define S1_    16        // K-splits for the fc batched GEMV
#define CPS1_  145       // ceil(2306/16)
#define S2_    32        // K-splits for the W1 GEMV
#define CPS2_  144       // (18432/4)/32 exactly

// ---------------------------------------------------------------------------
// Kernel 1: fc partials.  One wave per (channel, 16-row f-tile, K-split).
// D = A(16x4 of Wfc rows) x B(4x16 = x-chunk broadcast to all columns) + D.
// Every column of D equals the dot product; lane0/lane16 extract column 0.
// ---------------------------------------------------------------------------
__global__ __launch_bounds__(32)
void fc_wmma_kernel(const float* __restrict__ Wfc, const float* __restrict__ x,
                    float* __restrict__ partial1) {
  const int bid  = blockIdx.x;
  const int s    = bid & (S1_ - 1);
  const int mt   = (bid >> 4) & 3;
  const int c    = bid >> 6;
  const int lane = threadIdx.x;
  const int half = lane >> 4;               // 0: K0/K1 lanes, 1: K2/K3 lanes
  const int f    = mt * 16 + (lane & 15);

  const float* __restrict__ wrow = Wfc + (size_t)(c * F_ + f) * W_;
  const float* __restrict__ xr   = x   + (size_t)c * W_;

  int lo = s * CPS1_;
  int hi = lo + CPS1_; if (hi > NCHUNK) hi = NCHUNK;

  v8f acc0 = {};
  v8f acc1 = {};
  int ch = lo;
  for (; ch + 1 < hi; ch += 2) {            // wave-uniform trip count: EXEC stays full
    const int w0 = 4 * ch + 2 * half;
    v2f a0 = { wrow[w0],     wrow[w0 + 1] };
    v2f b0 = { xr[w0],       xr[w0 + 1]   };
    const int w1 = w0 + 4;
    v2f a1 = { wrow[w1],     wrow[w1 + 1] };
    v2f b1 = { xr[w1],       xr[w1 + 1]   };
    acc0 = __builtin_amdgcn_wmma_f32_16x16x4_f32(false, a0, false, b0,
                                                 (short)0, acc0, false, false);
    acc1 = __builtin_amdgcn_wmma_f32_16x16x4_f32(false, a1, false, b1,
                                                 (short)0, acc1, false, false);
  }
  if (ch < hi) {
    const int w0 = 4 * ch + 2 * half;
    v2f a0 = { wrow[w0], wrow[w0 + 1] };
    v2f b0 = { xr[w0],   xr[w0 + 1]   };
    acc0 = __builtin_amdgcn_wmma_f32_16x16x4_f32(false, a0, false, b0,
                                                 (short)0, acc0, false, false);
  }

  const int base = ((c * 4 + mt) * S1_ + s) * 16;
  if (lane == 0) {
#pragma unroll
    for (int r = 0; r < 8; ++r) partial1[base + r] = acc0[r] + acc1[r];
  } else if (lane == 16) {
#pragma unroll
    for (int r = 0; r < 8; ++r) partial1[base + 8 + r] = acc0[r] + acc1[r];
  }
}

// ---------------------------------------------------------------------------
// Kernel 2: 1D valid correlation (kernel height == channel count).
// x stays resident in the 192MB L2, so the strided channel loop is cheap.
// ---------------------------------------------------------------------------
__global__ __launch_bounds__(256)
void conv_kernel(const float* __restrict__ x, const float* __restrict__ Kc,
                 const float* __restrict__ bconv, float* __restrict__ cat) {
  const int t = blockIdx.x * blockDim.x + threadIdx.x;
  if (t >= TCONV_) return;
  float s = bconv[0];
  for (int c = 0; c < C_; ++c) {
    const float* __restrict__ xr = x  + (size_t)c * W_ + t;
    const float* __restrict__ kr = Kc + c * KW_;
#pragma unroll
    for (int k = 0; k < KW_; ++k) s += xr[k] * kr[k];
  }
  cat[TCONV_ + t] = s;
}

// ---------------------------------------------------------------------------
// Kernel 3: reduce fc partials + bias + the odd leftover column (w = 9224).
// ---------------------------------------------------------------------------
__global__ __launch_bounds__(256)
void fc_reduce_kernel(const float* __restrict__ partial1,
                      const float* __restrict__ Wfc, const float* __restrict__ x,
                      const float* __restrict__ bfc, float* __restrict__ cat) {
  const int i = blockIdx.x * blockDim.x + threadIdx.x;
  if (i >= C_ * F_) return;
  const int c  = i / F_;
  const int f  = i % F_;
  const int mt = f >> 4;
  const int m  = f & 15;
  float s = bfc[i];
  const int base = ((c * 4 + mt) * S1_) * 16 + m;
#pragma unroll
  for (int s1 = 0; s1 < S1_; ++s1) s += partial1[base + s1 * 16];
  s += Wfc[(size_t)i * W_ + (W_ - 1)] * x[(size_t)c * W_ + (W_ - 1)];
  cat[i] = s;
}

// ---------------------------------------------------------------------------
// Kernel 4: W1 @ cat partials, same WMMA GEMV scheme (8 m-tiles x 32 splits).
// ---------------------------------------------------------------------------
__global__ __launch_bounds__(32)
void gemv_wmma_kernel(const float* __restrict__ W1, const float* __restrict__ cat,
                      float* __restrict__ partial2) {
  const int bid  = blockIdx.x;
  const int s2   = bid & (S2_ - 1);
  const int mt   = bid >> 5;
  const int lane = threadIdx.x;
  const int half = lane >> 4;
  const int row  = mt * 16 + (lane & 15);

  const float* __restrict__ wrow = W1 + (size_t)row * CAT_;
  const int lo = s2 * CPS2_;

  v8f acc0 = {};
  v8f acc1 = {};
  for (int ch = 0; ch < CPS2_; ch += 2) {
    const int w0 = 4 * (lo + ch) + 2 * half;
    v2f a0 = { wrow[w0],    wrow[w0 + 1] };
    v2f b0 = { cat[w0],     cat[w0 + 1]  };
    const int w1 = w0 + 4;
    v2f a1 = { wrow[w1],    wrow[w1 + 1] };
    v2f b1 = { cat[w1],     cat[w1 + 1]  };
    acc0 = __builtin_amdgcn_wmma_f32_16x16x4_f32(false, a0, false, b0,
                                                 (short)0, acc0, false, false);
    acc1 = __builtin_amdgcn_wmma_f32_16x16x4_f32(false, a1, false, b1,
                                                 (short)0, acc1, false, false);
  }

  const int base = (mt * S2_ + s2) * 16;
  if (lane == 0) {
#pragma unroll
    for (int r = 0; r < 8; ++r) partial2[base + r] = acc0[r] + acc1[r];
  } else if (lane == 16) {
#pragma unroll
    for (int r = 0; r < 8; ++r) partial2[base + 8 + r] = acc0[r] + acc1[r];
  }
}

// ---------------------------------------------------------------------------
// Kernel 5: reduce W1 partials + b1, ReLU, dot with W2, ReLU, sigmoid.
// ---------------------------------------------------------------------------
__global__ __launch_bounds__(128)
void final_kernel(const float* __restrict__ partial2, const float* __restrict__ b1,
                  const float* __restrict__ W2, const float* __restrict__ b2,
                  float* __restrict__ out) {
  __shared__ float red[H1_];
  const int i  = threadIdx.x;
  const int mt = i >> 4;
  const int m  = i & 15;
  float s = b1[i];
#pragma unroll
  for (int s2 = 0; s2 < S2_; ++s2) s += partial2[(mt * S2_ + s2) * 16 + m];
  s = fmaxf(s, 0.0f);                       // second = relu(W1@cat + b1)
  red[i] = s * W2[i];
  __syncthreads();
  for (int off = H1_ / 2; off > 0; off >>= 1) {
    if (i < off) red[i] += red[i + off];
    __syncthreads();
  }
  if (i == 0) {
    float o = fmaxf(red[0] + b2[0], 0.0f);  // relu(W2@second + b2)
    out[0] = 1.0f / (1.0f + expf(-o));      // sigmoid
  }
}

// ---------------------------------------------------------------------------
extern "C" void kernel_launch(void* const* d_in, const int* in_sizes, int n_in,
                              void* d_out, int out_size, void* d_ws, size_t ws_size,
                              hipStream_t stream) {
  (void)in_sizes; (void)n_in; (void)out_size; (void)ws_size;
  const float* x     = (const float*)d_in[0];
  const float* Wfc   = (const float*)d_in[1];
  const float* bfc   = (const float*)d_in[2];
  const float* Kc    = (const float*)d_in[3];
  const float* bconv = (const float*)d_in[4];
  const float* W1    = (const float*)d_in[5];
  const float* b1    = (const float*)d_in[6];
  const float* W2    = (const float*)d_in[7];
  const float* b2    = (const float*)d_in[8];

  float* ws       = (float*)d_ws;
  float* partial1 = ws;                         // C*4*S1*16 = 147456 floats
  float* cat      = partial1 + C_ * 4 * S1_ * 16;  // 18432 floats
  float* partial2 = cat + CAT_;                 // 8*S2*16 = 4096 floats

  fc_wmma_kernel<<<C_ * 4 * S1_, 32, 0, stream>>>(Wfc, x, partial1);
  conv_kernel<<<(TCONV_ + 255) / 256, 256, 0, stream>>>(x, Kc, bconv, cat);
  fc_reduce_kernel<<<(C_ * F_ + 255) / 256, 256, 0, stream>>>(partial1, Wfc, x, bfc, cat);
  gemv_wmma_kernel<<<8 * S2_, 32, 0, stream>>>(W1, cat, partial2);
  final_kernel<<<1, H1_, 0, stream>>>(partial2, b1, W2, b2, (float*)d_out);
}